// DialogueRNNCell_70824010711207
// MI455X (gfx1250) — compile-verified
//
#include <hip/hip_runtime.h>
#include <hip/hip_bf16.h>
#include <math.h>

// Problem constants (match reference)
#define BB 4096
#define TT 128
#define PP 2
#define DD 256   // D_m = D_g = D_p = D_e
#define H3 768   // 3 * hidden

typedef float v2f __attribute__((ext_vector_type(2)));
typedef float v8f __attribute__((ext_vector_type(8)));

// row mapping: 0 -> b ; 1 -> b/2 (party broadcast) ; 2 -> b*2 + idx[b] (speaker gather)
__device__ __forceinline__ int maprow(int b, int mode, const int* __restrict__ idx) {
    if (mode == 0) return b;
    if (mode == 1) return b >> 1;
    return b * PP + idx[b];
}

// ---------------------------------------------------------------------------
// Attention: one workgroup (256 thr) per batch element b.
// Stage g_hist[:, b, :] (128x256 f32 = 128KB) into LDS once (single HBM pass
// over the 512MB tensor), compute scores, softmax over T, context, alpha.
// ---------------------------------------------------------------------------
__global__ void attn_kernel(const float* __restrict__ ghist,
                            const float* __restrict__ attn_w,
                            float* __restrict__ c_ws,
                            float* __restrict__ alpha_out) {
    const int b   = blockIdx.x;
    const int tid = threadIdx.x;
    extern __shared__ float sm[];
    float* gl  = sm;                 // [128][256]
    float* sc  = sm + TT * DD;       // [128] scores, later alpha
    float* esc = sc + TT;            // [128] exp(scores)

    // stage: coalesced float4 loads, contiguous per (t,b) row of 1KB
    for (int f = tid; f < (TT * DD) / 4; f += 256) {
        int t  = f >> 6;             // f / (256/4)
        int c4 = (f & 63) << 2;
        *(float4*)&gl[t * DD + c4] =
            *(const float4*)&ghist[((size_t)t * BB + b) * DD + c4];
    }
    const int lane = tid & 31;
    const int wv   = __builtin_amdgcn_readfirstlane(tid >> 5);  // scalar wave id
    float aw[8];
#pragma unroll
    for (int u = 0; u < 8; ++u) aw[u] = attn_w[lane * 8 + u];
    __syncthreads();

    // scores: each wave handles t = wv, wv+8, ... ; 8 elems/lane + xor-reduce
    for (int t = wv; t < TT; t += 8) {
        float s = 0.f;
#pragma unroll
        for (int u = 0; u < 8; ++u) s += gl[t * DD + lane * 8 + u] * aw[u];
        for (int o = 16; o > 0; o >>= 1) s += __shfl_xor(s, o, 32);
        if (lane == 0) sc[t] = s;
    }
    __syncthreads();

    // softmax over T (tiny: 128 values, redundant per-thread reduction is fine)
    if (tid < TT) {
        float m = -INFINITY;
        for (int t = 0; t < TT; ++t) m = fmaxf(m, sc[t]);
        esc[tid] = expf(sc[tid] - m);
    }
    __syncthreads();
    if (tid < TT) {
        float sum = 0.f;
        for (int t = 0; t < TT; ++t) sum += esc[t];
        float a = esc[tid] / sum;
        sc[tid] = a;                                   // reuse as alpha
        alpha_out[(size_t)b * TT + tid] = a;           // alpha[b,1,T]
    }
    __syncthreads();

    // context: c_[b, j] = sum_t alpha[t] * g[t, j]
    float acc = 0.f;
    for (int t = 0; t < TT; ++t) acc += sc[t] * gl[t * DD + tid];
    c_ws[(size_t)b * DD + tid] = acc;
}

// One 16x16 output tile accumulated over K with V_WMMA_F32_16X16X4_F32.
// Unroll-by-8 keeps 16 independent loads in flight per wait and issues
// 8 wmmas back-to-back.
template <int K>
__device__ __forceinline__ v8f tile_gemm(const float* __restrict__ arow,
                                         const float* __restrict__ wrow,
                                         v8f c) {
#pragma unroll 8
    for (int k = 0; k < K; k += 4) {
        v2f a  = *(const v2f*)&arow[k];
        v2f bb = *(const v2f*)&wrow[k];
        c = __builtin_amdgcn_wmma_f32_16x16x4_f32(
                false, a, false, bb, (short)0, c, false, false);
    }
    return c;
}

// ---------------------------------------------------------------------------
// Fused GRU cell via V_WMMA_F32_16X16X4_F32 (full fp32 fidelity).
//   gi = [X1|X2] @ Wih^T + bih   (K = KX, column-split over X1/X2)
//   gh = H @ Whh^T + bhh         (K = 256)
//   h' = (1-z)*n + z*h
// One workgroup = 16 batch rows, 8 waves, 96 16x16 output tiles (48 gi + 48 gh).
// A fragments from LDS (pad=4 floats: 16B-aligned float4 stage + bank-spread),
// B fragments streamed from L2-resident weights. gi/gh passes are separate
// loops with compile-time K so loop control is uniform/scalar.
// ---------------------------------------------------------------------------
template <int KX>
__global__ void gru_wmma_kernel(const float* __restrict__ X1,
                                const float* __restrict__ X2,
                                const float* __restrict__ Hin,
                                const float* __restrict__ Wih,
                                const float* __restrict__ Whh,
                                const float* __restrict__ bih,
                                const float* __restrict__ bhh,
                                const int*   __restrict__ idx,
                                float* __restrict__ Hout,
                                int x1mode, int x2mode) {
    extern __shared__ float sm[];
    const int tid = threadIdx.x;
    const int b0  = blockIdx.x * 16;
    constexpr int XSTR = KX + 4;                 // 16B aligned rows, bank shift 4
    float* xs  = sm;                             // [16][KX+4]
    float* hs  = xs + 16 * XSTR;                 // [16][260]
    float* gis = hs + 16 * 260;                  // [16][772]
    float* ghs = gis + 16 * 772;                 // [16][772]

    // ---- stage X tile (column-split X1/X2, gather/broadcast row modes) ----
    constexpr int NV4 = (16 * KX) >> 2;
    constexpr int KQ  = KX >> 2;
    for (int f = tid; f < NV4; f += 256) {
        int row = f / KQ;
        int col = (f % KQ) << 2;
        int b   = b0 + row;
        float4 v;
        if (col < DD || X2 == nullptr) {
            int r = maprow(b, x1mode, idx);
            v = *(const float4*)&X1[(size_t)r * DD + col];
        } else {
            int r = maprow(b, x2mode, idx);
            v = *(const float4*)&X2[(size_t)r * DD + (col - DD)];
        }
        *(float4*)&xs[row * XSTR + col] = v;
    }
    // ---- stage H tile ----
    for (int f = tid; f < (16 * DD) >> 2; f += 256) {
        int row = f >> 6;
        int col = (f & 63) << 2;
        *(float4*)&hs[row * 260 + col] =
            *(const float4*)&Hin[(size_t)(b0 + row) * DD + col];
    }
    __syncthreads();

    // ---- WMMA GEMM phase ----
    const int lane = tid & 31;
    const int wv   = __builtin_amdgcn_readfirstlane(tid >> 5);  // scalar
    const int half = lane >> 4;   // 0: lanes 0-15, 1: lanes 16-31
    const int l16  = lane & 15;

    // gi pass: K = KX, A from xs, B rows from Wih
    for (int t = wv; t < 48; t += 8) {
        const int n0 = t * 16;
        float bv = bih[n0 + l16];
        v8f c;
#pragma unroll
        for (int v = 0; v < 8; ++v) c[v] = bv;
        // A[m][k]: m = l16, k-pair = k + half*2 (LDS)
        // B[k][n]: n = n0 + l16 -> Wih[n][k + half*2] (global, L2-hot)
        c = tile_gemm<KX>(xs + l16 * XSTR + half * 2,
                          Wih + (size_t)(n0 + l16) * KX + half * 2, c);
#pragma unroll
        for (int v = 0; v < 8; ++v)
            gis[(v + 8 * half) * 772 + n0 + l16] = c[v];
    }
    // gh pass: K = 256, A from hs, B rows from Whh
    for (int t = wv; t < 48; t += 8) {
        const int n0 = t * 16;
        float bv = bhh[n0 + l16];
        v8f c;
#pragma unroll
        for (int v = 0; v < 8; ++v) c[v] = bv;
        c = tile_gemm<DD>(hs + l16 * 260 + half * 2,
                          Whh + (size_t)(n0 + l16) * DD + half * 2, c);
#pragma unroll
        for (int v = 0; v < 8; ++v)
            ghs[(v + 8 * half) * 772 + n0 + l16] = c[v];
    }
    __syncthreads();

    // ---- gate phase: thread tid owns column j across the 16 rows ----
    const int j = tid;
    for (int m = 0; m < 16; ++m) {
        float ir = gis[m * 772 + j];
        float iz = gis[m * 772 + DD + j];
        float in = gis[m * 772 + 2 * DD + j];
        float hr = ghs[m * 772 + j];
        float hz = ghs[m * 772 + DD + j];
        float hn = ghs[m * 772 + 2 * DD + j];
        float h  = hs[m * 260 + j];
        float r = 1.f / (1.f + expf(-(ir + hr)));
        float z = 1.f / (1.f + expf(-(iz + hz)));
        float n = tanhf(in + r * hn);
        Hout[(size_t)(b0 + m) * DD + j] = (1.f - z) * n + z * h;
    }
}

// q_ = q0 * (1-mask) + qs * mask  (speaker slot replaced)
__global__ void mask_kernel(const float* __restrict__ q0,
                            const float* __restrict__ qs,
                            const int*   __restrict__ idx,
                            float* __restrict__ qout) {
    int i = blockIdx.x * 256 + threadIdx.x;     // over B*P*D
    if (i >= BB * PP * DD) return;
    int b = i >> 9;              // / (P*D)
    int p = (i >> 8) & 1;        // party slot
    qout[i] = (p == idx[b]) ? qs[i] : q0[i];
}

extern "C" void kernel_launch(void* const* d_in, const int* in_sizes, int n_in,
                              void* d_out, int out_size, void* d_ws, size_t ws_size,
                              hipStream_t stream) {
    const float* U      = (const float*)d_in[0];
    const int*   qm_idx = (const int*)  d_in[1];
    const float* g_hist = (const float*)d_in[2];
    const float* q0     = (const float*)d_in[3];   // [B,P,D] == [B*P, D]
    const float* e0     = (const float*)d_in[4];
    const float* wg_ih  = (const float*)d_in[5];
    const float* wg_hh  = (const float*)d_in[6];
    const float* bg_ih  = (const float*)d_in[7];
    const float* bg_hh  = (const float*)d_in[8];
    const float* wp_ih  = (const float*)d_in[9];
    const float* wp_hh  = (const float*)d_in[10];
    const float* bp_ih  = (const float*)d_in[11];
    const float* bp_hh  = (const float*)d_in[12];
    const float* we_ih  = (const float*)d_in[13];
    const float* we_hh  = (const float*)d_in[14];
    const float* be_ih  = (const float*)d_in[15];
    const float* be_hh  = (const float*)d_in[16];
    const float* attn_w = (const float*)d_in[17];

    float* out       = (float*)d_out;
    float* out_g     = out;                          // [B, D]
    float* out_q     = out + (size_t)BB * DD;        // [B, P, D]
    float* out_e     = out + (size_t)BB * 3 * DD;    // [B, D]
    float* out_alpha = out + (size_t)BB * 4 * DD;    // [B, 1, T]

    float* ws    = (float*)d_ws;
    float* c_ws  = ws;                               // [B, D]
    float* qs_ws = ws + (size_t)BB * DD;             // [B*P, D]

    // 1) attention: context + alpha (single HBM pass over g_hist)
    {
        size_t smem = (size_t)(TT * DD + 2 * TT) * sizeof(float);
        attn_kernel<<<BB, 256, smem, stream>>>(g_hist, attn_w, c_ws, out_alpha);
    }
    // LDS size for GRU kernels: xs + hs + gi + gh
    auto gru_smem = [](int Kx) {
        return (size_t)(16 * (Kx + 4) + 16 * 260 + 2 * 16 * 772) * sizeof(float);
    };
    // 2) global GRU: x = [U | q0_sel], h = g_hist[T-1]
    gru_wmma_kernel<2 * DD><<<BB / 16, 256, gru_smem(2 * DD), stream>>>(
        U, q0, g_hist + (size_t)(TT - 1) * BB * DD,
        wg_ih, wg_hh, bg_ih, bg_hh, qm_idx, out_g,
        /*x1mode=*/0, /*x2mode=*/2);
    // 3) party GRU: N = B*P rows, x row = b/2 ([U | c_]), h = q0 flat
    gru_wmma_kernel<2 * DD><<<(BB * PP) / 16, 256, gru_smem(2 * DD), stream>>>(
        U, c_ws, q0,
        wp_ih, wp_hh, bp_ih, bp_hh, qm_idx, qs_ws,
        /*x1mode=*/1, /*x2mode=*/1);
    // 4) merge q_ (listeners keep q0, speaker slot takes qs_)
    mask_kernel<<<(BB * PP * DD) / 256, 256, 0, stream>>>(q0, qs_ws, qm_idx, out_q);
    // 5) emotion GRU: x = q_[b, qm_idx[b]], h = e0
    gru_wmma_kernel<DD><<<BB / 16, 256, gru_smem(DD), stream>>>(
        out_q, nullptr, e0,
        we_ih, we_hh, be_ih, be_hh, qm_idx, out_e,
        /*x1mode=*/2, /*x2mode=*/0);
}